// SurpriseBasedMemory_62972810494565
// MI455X (gfx1250) — compile-verified
//
#include <hip/hip_runtime.h>
#include <cfloat>
#include <math.h>

// ---- problem constants (from reference setup_inputs) ----
#define TOK   4096      // B*S = 2*2048
#define HID   1024      // H
#define MEMN  4096      // M
#define KUPD  64        // MAX_UPDATES
#define THRESH 3e-7f    // SURPRISE_THRESHOLD

typedef __attribute__((ext_vector_type(2))) float    v2f;
typedef __attribute__((ext_vector_type(8))) float    v8f;
typedef __attribute__((ext_vector_type(4))) unsigned v4u;
typedef __attribute__((ext_vector_type(8))) unsigned v8u;

// ---------------- block reductions (256 threads) ----------------
__device__ __forceinline__ float blk_sum(float v, float* red) {
  const int tid = threadIdx.x;
  red[tid] = v; __syncthreads();
  #pragma unroll
  for (int off = 128; off > 0; off >>= 1) {
    if (tid < off) red[tid] += red[tid + off];
    __syncthreads();
  }
  float r = red[0]; __syncthreads();
  return r;
}
__device__ __forceinline__ float blk_max(float v, float* red) {
  const int tid = threadIdx.x;
  red[tid] = v; __syncthreads();
  #pragma unroll
  for (int off = 128; off > 0; off >>= 1) {
    if (tid < off) red[tid] = fmaxf(red[tid], red[tid + off]);
    __syncthreads();
  }
  float r = red[0]; __syncthreads();
  return r;
}

// ---------------- TDM: async 2-D tile load global -> LDS ----------------
// Builds a Tensor DMA Descriptor (D#, groups 0/1 per CDNA5 ISA 8.3/8.4) and
// issues tensor_load_to_lds (2-group form, <=2D tensor). Tracked by TENSORcnt.
// padIntLog: pad after every (2<<padIntLog) DWORDs; padAmtM1: pad (padAmtM1+1) DWORDs.
__device__ __forceinline__ void tdm_load_2d(
    unsigned ldsByte, const float* g, unsigned tileW, unsigned tileH,
    unsigned strideElems, unsigned padIntLog, unsigned padAmtM1)
{
  const unsigned long long ga = (unsigned long long)g;
  v4u g0;
  g0.x = 1u;                                        // count=1 (valid user D#)
  g0.y = ldsByte;                                   // lds_addr
  g0.z = (unsigned)ga;                              // global_addr[31:0]
  g0.w = ((unsigned)(ga >> 32) & 0x01ffffffu)       // global_addr[56:32]
       | (2u << 30);                                // type=2 ("image")
  v8u g1;
  g1[0] = (2u << 16)                                // data_size = 4B
        | (1u << 20)                                // pad_enable
        | (padIntLog << 22) | (padAmtM1 << 25);
  g1[1] = 0u;                                       // abar=0; tensor_dim0[15:0]=0
  g1[2] = 0x4000u;                                  // tensor_dim0 = 1<<30 (no OOB)
  g1[3] = 0x4000u | (tileW << 16);                  // tensor_dim1 = 1<<30; tile_dim0
  g1[4] = tileH;                                    // tile_dim1; tile_dim2 = 0
  g1[5] = strideElems;                              // tensor_dim0_stride[31:0]
  g1[6] = 0u;                                       // stride hi; dim1_stride lo
  g1[7] = 0u;
  asm volatile("tensor_load_to_lds %0, %1" :: "s"(g0), "s"(g1) : "memory");
}

// ---------------- fp32 WMMA GEMM, TDM-staged, double-buffered ----------------
// D[M,N] = alpha * (A[M,K] @ opB) + bias[N] + beta * Res[M,N]
// opB = B[K,N] (TRANSB=false) or B[N,K]^T (TRANSB=true)
// Block tile 128x64, K-chunk 32. 8 waves, each computes a 32x32 quadrant
// (2x2 subtiles of 16x16 -> 4 WMMA accumulators, 4 WMMAs per 4 frag loads).
template<bool TRANSB>
__global__ __launch_bounds__(256) void gemm_wmma_f32(
    const float* __restrict__ A, const float* __restrict__ B,
    float* __restrict__ D,
    const float* __restrict__ bias, const float* __restrict__ Res,
    float alpha, float beta, int N, int K)
{
  constexpr int BROWS = TRANSB ? 64 : 32;   // Bt[n][k] rows=64 | B[k][n] rows=32
  constexpr int BSTR  = TRANSB ? 36 : 68;   // padded row strides (TDM pad)
  __shared__ float As[2][128][36];
  __shared__ float Bs[2][BROWS][BSTR];

  const int tid  = threadIdx.x;
  const int lane = tid & 31;
  const int wave = tid >> 5;
  const int fr   = lane & 15;    // fragment row (M for A, N for B)
  const int fp   = lane >> 4;    // lane half -> K-pair selector
  const int rowBase = blockIdx.y << 7;
  const int colBase = blockIdx.x << 6;
  const int r0 = (wave & 3) << 5;    // wave's 32-row quadrant
  const int c0 = (wave >> 2) << 5;   // wave's 32-col quadrant

  const unsigned ldsA[2] = { (unsigned)(unsigned long long)(void*)&As[0][0][0],
                             (unsigned)(unsigned long long)(void*)&As[1][0][0] };
  const unsigned ldsB[2] = { (unsigned)(unsigned long long)(void*)&Bs[0][0][0],
                             (unsigned)(unsigned long long)(void*)&Bs[1][0][0] };

  v8f acc00 = {}, acc01 = {}, acc10 = {}, acc11 = {};
  const int nChunks = K >> 5;

  // prologue: wave 0 DMAs chunk 0 into buffer 0
  if (wave == 0) {
    tdm_load_2d(ldsA[0], A + (size_t)rowBase * K, 32u, 128u, (unsigned)K, 4u, 3u);
    if (TRANSB)
      tdm_load_2d(ldsB[0], B + (size_t)colBase * K, 32u, 64u, (unsigned)K, 4u, 3u);
    else
      tdm_load_2d(ldsB[0], B + colBase, 64u, 32u, (unsigned)N, 5u, 3u);
  }

  for (int ch = 0; ch < nChunks; ++ch) {
    const int cur = ch & 1;
    if (wave == 0) {
      if (ch + 1 < nChunks) {           // prefetch next chunk into other buffer
        const int k0 = (ch + 1) << 5;
        tdm_load_2d(ldsA[cur ^ 1], A + (size_t)rowBase * K + k0, 32u, 128u,
                    (unsigned)K, 4u, 3u);
        if (TRANSB)
          tdm_load_2d(ldsB[cur ^ 1], B + (size_t)colBase * K + k0, 32u, 64u,
                      (unsigned)K, 4u, 3u);
        else
          tdm_load_2d(ldsB[cur ^ 1], B + (size_t)k0 * N + colBase, 64u, 32u,
                      (unsigned)N, 5u, 3u);
        __builtin_amdgcn_s_wait_tensorcnt(2);   // chunk ch landed
      } else {
        __builtin_amdgcn_s_wait_tensorcnt(0);
      }
    }
    __syncthreads();

    // A frag: VGPR0 = K{0|2}, VGPR1 = K{1|3} per lane half -> single b64 read
    #pragma unroll
    for (int kk = 0; kk < 8; ++kk) {
      const int kc = (kk << 2) + (fp << 1);
      v2f a0 = *(const v2f*)&As[cur][r0 + fr][kc];
      v2f a1 = *(const v2f*)&As[cur][r0 + 16 + fr][kc];
      v2f b0, b1;
      if (TRANSB) {
        b0 = *(const v2f*)&Bs[cur][c0 + fr][kc];
        b1 = *(const v2f*)&Bs[cur][c0 + 16 + fr][kc];
      } else {
        b0.x = Bs[cur][kc][c0 + fr];      b0.y = Bs[cur][kc + 1][c0 + fr];
        b1.x = Bs[cur][kc][c0 + 16 + fr]; b1.y = Bs[cur][kc + 1][c0 + 16 + fr];
      }
      acc00 = __builtin_amdgcn_wmma_f32_16x16x4_f32(false, a0, false, b0, (short)0, acc00, false, false);
      acc01 = __builtin_amdgcn_wmma_f32_16x16x4_f32(false, a0, false, b1, (short)0, acc01, false, false);
      acc10 = __builtin_amdgcn_wmma_f32_16x16x4_f32(false, a1, false, b0, (short)0, acc10, false, false);
      acc11 = __builtin_amdgcn_wmma_f32_16x16x4_f32(false, a1, false, b1, (short)0, acc11, false, false);
    }
    __syncthreads();
  }

  // epilogue: C/D layout -> lane 0-15: M=vgpr, lane 16-31: M=vgpr+8, N=fr
  const v8f accs[2][2] = { { acc00, acc01 }, { acc10, acc11 } };
  #pragma unroll
  for (int rr = 0; rr < 2; ++rr) {
    #pragma unroll
    for (int cc = 0; cc < 2; ++cc) {
      const int col = colBase + c0 + (cc << 4) + fr;
      const float bv = bias ? bias[col] : 0.0f;
      #pragma unroll
      for (int v = 0; v < 8; ++v) {
        const int row = rowBase + r0 + (rr << 4) + v + (fp << 3);
        float val = accs[rr][cc][v] * alpha + bv;
        if (Res) val = fmaf(beta, Res[(size_t)row * N + col], val);
        D[(size_t)row * N + col] = val;
      }
    }
  }
}

// ---------------- LayerNorm: one block per token ----------------
__global__ __launch_bounds__(256) void ln_kernel(
    const float* __restrict__ x, const float* __restrict__ g,
    const float* __restrict__ b, float* __restrict__ xn)
{
  __shared__ float red[256];
  const int row = blockIdx.x;
  const float* xr = x + (size_t)row * HID;
  float lv[4]; float s = 0.f;
  #pragma unroll
  for (int i = 0; i < 4; ++i) { lv[i] = xr[threadIdx.x + (i << 8)]; s += lv[i]; }
  const float mu = blk_sum(s, red) * (1.0f / HID);
  float var = 0.f;
  #pragma unroll
  for (int i = 0; i < 4; ++i) { float d = lv[i] - mu; var += d * d; }
  var = blk_sum(var, red) * (1.0f / HID);
  const float inv = rsqrtf(var + 1e-12f);
  #pragma unroll
  for (int i = 0; i < 4; ++i) {
    const int j = threadIdx.x + (i << 8);
    xn[(size_t)row * HID + j] = (lv[i] - mu) * inv * g[j] + b[j];
  }
}

// ---------------- row softmax over M=4096, in place ----------------
__global__ __launch_bounds__(256) void softmax_kernel(float* __restrict__ S)
{
  __shared__ float red[256];
  const size_t base = (size_t)blockIdx.x * MEMN;
  float lv[16]; float mx = -FLT_MAX;
  #pragma unroll
  for (int i = 0; i < 16; ++i) { lv[i] = S[base + threadIdx.x + (i << 8)]; mx = fmaxf(mx, lv[i]); }
  mx = blk_max(mx, red);
  float sum = 0.f;
  #pragma unroll
  for (int i = 0; i < 16; ++i) { lv[i] = expf(lv[i] - mx); sum += lv[i]; }
  const float inv = 1.0f / blk_sum(sum, red);
  #pragma unroll
  for (int i = 0; i < 16; ++i) S[base + threadIdx.x + (i << 8)] = lv[i] * inv;
}

// ---------------- softmax backward row op: A := A*(G - sum(A*G)) ----------------
__global__ __launch_bounds__(256) void dlogit_kernel(
    float* __restrict__ A1, const float* __restrict__ G)
{
  __shared__ float red[256];
  const size_t base = (size_t)blockIdx.x * MEMN;
  float la[16], lg[16]; float t = 0.f;
  #pragma unroll
  for (int i = 0; i < 16; ++i) {
    la[i] = A1[base + threadIdx.x + (i << 8)];
    lg[i] = G [base + threadIdx.x + (i << 8)];
    t += la[i] * lg[i];
  }
  t = blk_sum(t, red);
  #pragma unroll
  for (int i = 0; i < 16; ++i) A1[base + threadIdx.x + (i << 8)] = la[i] * (lg[i] - t);
}

// ---------------- elementwise: o = a - b ----------------
__global__ __launch_bounds__(256) void sub_kernel(
    const float* __restrict__ a, const float* __restrict__ b,
    float* __restrict__ o, int n)
{
  const int i = blockIdx.x * 256 + threadIdx.x;
  if (i < n) o[i] = a[i] - b[i];
}

// ---------------- surprise = mean |g| per token row ----------------
__global__ __launch_bounds__(256) void surprise_kernel(
    const float* __restrict__ gx, float* __restrict__ sp)
{
  __shared__ float red[256];
  const size_t base = (size_t)blockIdx.x * HID;
  float s = 0.f;
  #pragma unroll
  for (int i = 0; i < 4; ++i) s += fabsf(gx[base + threadIdx.x + (i << 8)]);
  s = blk_sum(s, red);
  if (threadIdx.x == 0) sp[blockIdx.x] = s * (1.0f / HID);
}

// ---------------- top-64 surprise -> 64 least-important slots, scatter xn rows ----
__global__ __launch_bounds__(256) void topk_update_kernel(
    const float* __restrict__ sp, const float* __restrict__ imp,
    const float* __restrict__ xn, float* __restrict__ newmem)
{
  __shared__ float vals[TOK];            // reused: masked surprise, then importance
  __shared__ float rv[256]; __shared__ int ri[256];
  __shared__ int   top_idx[KUPD]; __shared__ float top_val[KUPD];
  __shared__ int   slot_idx[KUPD];
  const int tid = threadIdx.x;

  // masked surprise (mirror jnp.where(flat_s > THR, flat_s, -inf))
  #pragma unroll
  for (int i = 0; i < 16; ++i) {
    const int j = tid + (i << 8);
    const float s = sp[j];
    vals[j] = (s > THRESH) ? s : -FLT_MAX;
  }
  __syncthreads();
  // iterative argmax, stable ties (lowest index first) like lax.top_k
  for (int k = 0; k < KUPD; ++k) {
    float bv = -FLT_MAX; int bi = TOK;
    #pragma unroll
    for (int i = 0; i < 16; ++i) {
      const int j = tid + (i << 8);
      const float v = vals[j];
      if (v > bv || (v == bv && j < bi)) { bv = v; bi = j; }
    }
    rv[tid] = bv; ri[tid] = bi; __syncthreads();
    for (int off = 128; off > 0; off >>= 1) {
      if (tid < off) {
        const float ov = rv[tid + off]; const int oi = ri[tid + off];
        if (ov > rv[tid] || (ov == rv[tid] && oi < ri[tid])) { rv[tid] = ov; ri[tid] = oi; }
      }
      __syncthreads();
    }
    if (tid == 0) { top_idx[k] = ri[0]; top_val[k] = rv[0]; vals[ri[0]] = -FLT_MAX; }
    __syncthreads();
  }
  // 64 least-important slots (top_k of -importance)
  #pragma unroll
  for (int i = 0; i < 16; ++i) { const int j = tid + (i << 8); vals[j] = imp[j]; }
  __syncthreads();
  for (int k = 0; k < KUPD; ++k) {
    float bv = FLT_MAX; int bi = TOK;
    #pragma unroll
    for (int i = 0; i < 16; ++i) {
      const int j = tid + (i << 8);
      const float v = vals[j];
      if (v < bv || (v == bv && j < bi)) { bv = v; bi = j; }
    }
    rv[tid] = bv; ri[tid] = bi; __syncthreads();
    for (int off = 128; off > 0; off >>= 1) {
      if (tid < off) {
        const float ov = rv[tid + off]; const int oi = ri[tid + off];
        if (ov < rv[tid] || (ov == rv[tid] && oi < ri[tid])) { rv[tid] = ov; ri[tid] = oi; }
      }
      __syncthreads();
    }
    if (tid == 0) { slot_idx[k] = ri[0]; vals[ri[0]] = FLT_MAX; }
    __syncthreads();
  }
  // scatter valid rows of xn into new_mem (whole block copies one row)
  for (int k = 0; k < KUPD; ++k) {
    if (top_val[k] > THRESH) {
      const int src = top_idx[k], dst = slot_idx[k];
      #pragma unroll
      for (int i = 0; i < 4; ++i) {
        const int j = tid + (i << 8);
        newmem[(size_t)dst * HID + j] = xn[(size_t)src * HID + j];
      }
    }
  }
}

// ---------------- host orchestration ----------------
extern "C" void kernel_launch(void* const* d_in, const int* in_sizes, int n_in,
                              void* d_out, int out_size, void* d_ws, size_t ws_size,
                              hipStream_t stream) {
  (void)in_sizes; (void)n_in; (void)out_size; (void)ws_size;
  const float* hidden = (const float*)d_in[0];
  const float* ln_g   = (const float*)d_in[1];
  const float* ln_b   = (const float*)d_in[2];
  const float* Wq     = (const float*)d_in[3];
  const float* bq     = (const float*)d_in[4];
  const float* Wo     = (const float*)d_in[5];
  const float* bo     = (const float*)d_in[6];
  const float* mem    = (const float*)d_in[7];
  const float* imp    = (const float*)d_in[8];
  // d_in[9] (usage) does not affect the output
  float* out = (float*)d_out;

  const size_t TH_ = (size_t)TOK * HID;
  const size_t TM_ = (size_t)TOK * MEMN;
  float* ws   = (float*)d_ws;
  float* xn   = ws;            // [T,H]
  float* q    = xn + TH_;      // [T,H]
  float* attn = q + TH_;       // [T,M]  S1 -> A1 -> dlogit
  float* gb   = attn + TM_;    // [T,M]  G   -> S2 -> A2
  float* r2   = gb + TM_;      // [T,H]  mo  -> r2 = xn - mo
  float* dq   = r2 + TH_;      // [T,H]  dq  -> mo2
  float* gx   = dq + TH_;      // [T,H]
  float* nm   = gx + TH_;      // [M,H]  new_mem
  float* sp   = nm + TH_;      // [T]

  const float cgrad = 2.0f / ((float)TOK * (float)HID);   // d(mean sq)/dx scale
  dim3 blk(256);
  const dim3 gH(HID / 64, TOK / 128);    // GEMMs with N=H
  const dim3 gM(MEMN / 64, TOK / 128);   // GEMMs with N=M

  // 1) x-normalized
  ln_kernel<<<TOK, blk, 0, stream>>>(hidden, ln_g, ln_b, xn);
  // 2) q = xn @ Wq + bq
  gemm_wmma_f32<false><<<gH, blk, 0, stream>>>(xn, Wq, q, bq, nullptr, 1.f, 0.f, HID, HID);
  // 3) S1 = q @ mem^T ; A1 = softmax(S1)
  gemm_wmma_f32<true><<<gM, blk, 0, stream>>>(q, mem, attn, nullptr, nullptr, 1.f, 0.f, MEMN, HID);
  softmax_kernel<<<TOK, blk, 0, stream>>>(attn);
  // 4) mo = A1 @ mem ; r2 = xn - mo
  gemm_wmma_f32<false><<<gH, blk, 0, stream>>>(attn, mem, r2, nullptr, nullptr, 1.f, 0.f, HID, MEMN);
  sub_kernel<<<(int)(TH_ / 256), blk, 0, stream>>>(xn, r2, r2, (int)TH_);
  // 5) dattn = -c * (r2 @ mem^T) ; dlogit = A1*(dattn - rowsum(A1*dattn))
  gemm_wmma_f32<true><<<gM, blk, 0, stream>>>(r2, mem, gb, nullptr, nullptr, -cgrad, 0.f, MEMN, HID);
  dlogit_kernel<<<TOK, blk, 0, stream>>>(attn, gb);
  // 6) dq = dlogit @ mem ; gx = dq @ Wq^T + c*r2 ; surprise = mean|gx|
  gemm_wmma_f32<false><<<gH, blk, 0, stream>>>(attn, mem, dq, nullptr, nullptr, 1.f, 0.f, HID, MEMN);
  gemm_wmma_f32<true><<<gH, blk, 0, stream>>>(dq, Wq, gx, nullptr, r2, 1.f, cgrad, HID, HID);
  surprise_kernel<<<TOK, blk, 0, stream>>>(gx, sp);
  // 7) new_mem = mem with top-64 surprise tokens scattered into 64 cheapest slots
  hipMemcpyAsync(nm, mem, TH_ * sizeof(float), hipMemcpyDeviceToDevice, stream);
  topk_update_kernel<<<1, blk, 0, stream>>>(sp, imp, xn, nm);
  // 8) re-query: A2 = softmax(q @ nm^T) ; mo2 = A2 @ nm
  gemm_wmma_f32<true><<<gM, blk, 0, stream>>>(q, nm, gb, nullptr, nullptr, 1.f, 0.f, MEMN, HID);
  softmax_kernel<<<TOK, blk, 0, stream>>>(gb);
  gemm_wmma_f32<false><<<gH, blk, 0, stream>>>(gb, nm, dq, nullptr, nullptr, 1.f, 0.f, HID, MEMN);
  // 9) out = hidden + mo2 @ Wo + bo
  gemm_wmma_f32<false><<<gH, blk, 0, stream>>>(dq, Wo, out, bo, hidden, 1.f, 1.f, HID, HID);
}